// PagedAttention_83021717832328
// MI455X (gfx1250) — compile-verified
//
#include <hip/hip_runtime.h>

typedef __attribute__((ext_vector_type(2))) float v2f;
typedef __attribute__((ext_vector_type(4))) float v4f;
typedef __attribute__((ext_vector_type(8))) float v8f;
typedef __attribute__((ext_vector_type(4))) int   v4i;

#define B_      16
#define H_      32
#define KVH_    8
#define G_      4
#define D_      128
#define T_      4096
#define NSPLIT  32
#define SPLIT_TOK (T_ / NSPLIT)   /* 128 tokens per split */
#define CH      16                /* chunk of tokens per WMMA tile */
#define KSTRIDE 132               /* 128 + 4 pad -> conflict-free column reads */
#define PSTRIDE 20
#define SCALE   0.08838834764831845f  /* 1/sqrt(128) */
#define NEG_BIG (-1e30f)
#define ROPE_C  (-0.2076205059304601f) /* -log2(10000)/64 */

// CDNA5 async memory->LDS path (ASYNCcnt) if the toolchain exposes it.
#if defined(__HIP_DEVICE_COMPILE__) && __has_builtin(__builtin_amdgcn_global_load_async_to_lds_b128)
#define USE_ASYNC_LDS 1
#else
#define USE_ASYNC_LDS 0
#endif

__device__ __forceinline__ float fast_exp2(float x) {
  return __builtin_amdgcn_exp2f(x);   // v_exp_f32
}

__device__ __forceinline__ v8f wmma4(v2f a, v2f b, v8f c) {
  // D = A(16x4 f32) * B(4x16 f32) + C(16x16 f32)
  return __builtin_amdgcn_wmma_f32_16x16x4_f32(
      /*neg_a=*/false, a, /*neg_b=*/false, b,
      /*c_mod=*/(short)0, c, /*reuse_a=*/false, /*reuse_b=*/false);
}

__device__ __forceinline__ float rmax16(float x) {
  x = fmaxf(x, __shfl_xor(x, 8, 16));
  x = fmaxf(x, __shfl_xor(x, 4, 16));
  x = fmaxf(x, __shfl_xor(x, 2, 16));
  x = fmaxf(x, __shfl_xor(x, 1, 16));
  return x;
}
__device__ __forceinline__ float rsum16(float x) {
  x += __shfl_xor(x, 8, 16);
  x += __shfl_xor(x, 4, 16);
  x += __shfl_xor(x, 2, 16);
  x += __shfl_xor(x, 1, 16);
  return x;
}

// ---------------------------------------------------------------------------
// Kernel 1: one wave per (b, kv-head, split). Flash-decoding over 128 tokens.
// ---------------------------------------------------------------------------
__global__ __launch_bounds__(32) void pa_split_kernel(
    const float* __restrict__ query, const float* __restrict__ k_cache,
    const float* __restrict__ v_cache, const long long* __restrict__ slot_tables,
    const int* __restrict__ context_lens, float* __restrict__ ws) {
  __shared__ __align__(16) float Qs[G_][D_];        // roped, scale-folded Q
  __shared__ __align__(16) float Kc[CH][KSTRIDE];   // staged K chunk (roped in place)
  __shared__ __align__(16) float Vc[CH][KSTRIDE];   // staged V chunk
  __shared__ __align__(16) float Ps[G_][PSTRIDE];   // softmax probs (C->A relayout)

  const int L    = threadIdx.x;        // lane 0..31
  const int m16  = L & 15;
  const int half = L >> 4;             // 0 | 1 (upper half-wave)
  const int bk   = blockIdx.x;         // 0..127
  const int b    = bk >> 3;
  const int kvh  = bk & 7;
  const int split = blockIdx.y;
  const int ctx  = context_lens[b];

  // ---- RoPE the 4 query rows of this kv-head group into LDS (scale folded in)
  {
    const int pos = ctx - 1;
    const float* qb = query + ((size_t)b * H_ + (size_t)kvh * G_) * D_;
#pragma unroll
    for (int it = 0; it < 8; ++it) {
      int idx = it * 32 + L;           // 0..255 = 4 rows * 64 pairs
      int g = idx >> 6, j = idx & 63;
      float x0 = qb[g * D_ + j];
      float x1 = qb[g * D_ + j + 64];
      float ang = (float)pos * fast_exp2((float)j * ROPE_C);
      float cs = __cosf(ang), sn = __sinf(ang);
      Qs[g][j]      = SCALE * (x0 * cs - x1 * sn);
      Qs[g][j + 64] = SCALE * (x1 * cs + x0 * sn);
    }
  }
  __syncthreads();

  // ---- Pull Q into WMMA A-operand registers: 32 tiles over D, K=4 each.
  // A 16x4 f32 layout: lane -> M=L%16, VGPR pair -> K = 2*half + {0,1}.
  v2f qa[32];
#pragma unroll
  for (int dt = 0; dt < 32; ++dt) {
    v2f a = {0.0f, 0.0f};
    if (m16 < G_) a = *(const v2f*)&Qs[m16][dt * 4 + 2 * half];
    qa[dt] = a;
  }

  const int tbeg = split * SPLIT_TOK;
  const int tend = min(ctx, tbeg + SPLIT_TOK);

  float mrun[G_], lrun[G_];
#pragma unroll
  for (int g = 0; g < G_; ++g) { mrun[g] = NEG_BIG; lrun[g] = 0.0f; }
  v8f oacc[8];
#pragma unroll
  for (int t = 0; t < 8; ++t) oacc[t] = (v8f){0, 0, 0, 0, 0, 0, 0, 0};

  for (int cbase = tbeg; cbase < tend; cbase += CH) {
    // -- slots for this 16-token chunk (lanes 16-31 mirror 0-15)
    const int tok = cbase + m16;
    const int slot = (int)slot_tables[(size_t)b * T_ + tok];
    if (cbase + CH < T_)
      __builtin_prefetch(&slot_tables[(size_t)b * T_ + cbase + CH], 0, 1);

    // -- stage K/V rows: one fully-coalesced 512B row per iteration
#pragma unroll
    for (int r = 0; r < CH; ++r) {
      int srow = __builtin_amdgcn_readlane(slot, r);
      const float* kr = k_cache + ((size_t)srow * KVH_ + kvh) * D_ + 4 * L;
      const float* vr = v_cache + ((size_t)srow * KVH_ + kvh) * D_ + 4 * L;
#if USE_ASYNC_LDS
      __builtin_amdgcn_global_load_async_to_lds_b128(
          (v4i*)kr, (v4i*)&Kc[r][4 * L], /*offset=*/0, /*cpol=*/0);
      __builtin_amdgcn_global_load_async_to_lds_b128(
          (v4i*)vr, (v4i*)&Vc[r][4 * L], /*offset=*/0, /*cpol=*/0);
#else
      v4f kq = *(const v4f*)kr;
      v4f vq = *(const v4f*)vr;
      *(v4f*)&Kc[r][4 * L] = kq;
      *(v4f*)&Vc[r][4 * L] = vq;
#endif
    }
#if USE_ASYNC_LDS
#if __has_builtin(__builtin_amdgcn_s_wait_asynccnt)
    __builtin_amdgcn_s_wait_asynccnt(0);
#else
    asm volatile("s_wait_asynccnt 0" ::: "memory");
#endif
#endif
    __syncthreads();

    // -- RoPE K rows in LDS (position = table index cbase+r)
#pragma unroll 4
    for (int it = 0; it < 32; ++it) {
      int idx = it * 32 + L;           // 0..1023 = 16 rows * 64 pairs
      int r = idx >> 6, j = idx & 63;
      float x0 = Kc[r][j], x1 = Kc[r][j + 64];
      float ang = (float)(cbase + r) * fast_exp2((float)j * ROPE_C);
      float cs = __cosf(ang), sn = __sinf(ang);
      Kc[r][j]      = x0 * cs - x1 * sn;
      Kc[r][j + 64] = x1 * cs + x0 * sn;
    }
    __syncthreads();

    // -- S = Q * K^T  (32 WMMAs over D). B 4x16 layout: lane -> N=L%16 (token),
    //    VGPR v -> K-row v + 2*half, i.e. 2 consecutive d -> one b64 LDS read.
    v8f sc = (v8f){0, 0, 0, 0, 0, 0, 0, 0};
#pragma unroll
    for (int dt = 0; dt < 32; ++dt) {
      v2f bk_ = *(const v2f*)&Kc[m16][dt * 4 + 2 * half];
      sc = wmma4(qa[dt], bk_, sc);
    }

    // -- online softmax per query row (token = lane % 16 in C layout)
    const bool valid = (cbase + m16) < tend;
    float pv[G_];
#pragma unroll
    for (int g = 0; g < G_; ++g) {
      float s = valid ? sc[g] : NEG_BIG;
      float mnew = fmaxf(mrun[g], rmax16(s));
      float p = valid ? __expf(s - mnew) : 0.0f;
      float rs = rsum16(p);
      float alpha = __expf(mrun[g] - mnew);
      lrun[g] = lrun[g] * alpha + rs;
      mrun[g] = mnew;
      pv[g] = p;
#pragma unroll
      for (int t = 0; t < 8; ++t) oacc[t][g] = oacc[t][g] * alpha;
    }
    if (L < 16) {
#pragma unroll
      for (int g = 0; g < G_; ++g) Ps[g][L] = pv[g];
    }
    __syncthreads();

    // -- O += P * V  (4 token sub-tiles x 8 D-tiles = 32 WMMAs)
#pragma unroll
    for (int s4 = 0; s4 < 4; ++s4) {
      v2f aP = {0.0f, 0.0f};
      if (m16 < G_) aP = *(const v2f*)&Ps[m16][s4 * 4 + 2 * half];
      const int r0 = s4 * 4 + 2 * half;
#pragma unroll
      for (int d1 = 0; d1 < 8; ++d1) {
        v2f bV;
        bV.x = Vc[r0][d1 * 16 + m16];
        bV.y = Vc[r0 + 1][d1 * 16 + m16];
        oacc[d1] = wmma4(aP, bV, oacc[d1]);
      }
    }
    __syncthreads();
  }

  // -- write partials: [acc(128) | m | l] per (bk, split, g)
  float* wsb = ws + (((size_t)bk * NSPLIT + split) * G_) * (D_ + 2);
#pragma unroll
  for (int g = 0; g < G_; ++g) {
    float* row = wsb + g * (D_ + 2);
    if (L < 16) {
#pragma unroll
      for (int d1 = 0; d1 < 8; ++d1) row[d1 * 16 + L] = oacc[d1][g];
    }
    if (L == 0) { row[D_] = mrun[g]; row[D_ + 1] = lrun[g]; }
  }
}

// ---------------------------------------------------------------------------
// Kernel 2: merge the NSPLIT partials (log-sum-exp combine) -> d_out
// ---------------------------------------------------------------------------
__global__ __launch_bounds__(128) void pa_combine_kernel(
    const float* __restrict__ ws, float* __restrict__ out) {
  const int d   = threadIdx.x;       // 0..127
  const int idx = blockIdx.x;        // 0..511 = bk*G + g
  const int bk  = idx >> 2, g = idx & 3;
  const int b   = bk >> 3, kvh = bk & 7;
  const float* base = ws + ((size_t)bk * NSPLIT * G_ + g) * (D_ + 2);
  const int sstride = G_ * (D_ + 2);

  float M = NEG_BIG;
  for (int s = 0; s < NSPLIT; ++s) M = fmaxf(M, base[s * sstride + D_]);
  float num = 0.0f, den = 0.0f;
  for (int s = 0; s < NSPLIT; ++s) {
    const float* row = base + s * sstride;
    float w = __expf(row[D_] - M);
    num = fmaf(w, row[d], num);
    den = fmaf(w, row[D_ + 1], den);
  }
  out[((size_t)b * H_ + (size_t)(kvh * G_ + g)) * D_ + d] = num / den;
}

extern "C" void kernel_launch(void* const* d_in, const int* in_sizes, int n_in,
                              void* d_out, int out_size, void* d_ws, size_t ws_size,
                              hipStream_t stream) {
  (void)in_sizes; (void)n_in; (void)out_size; (void)ws_size;
  const float*      query        = (const float*)d_in[0];
  const float*      k_cache      = (const float*)d_in[1];
  const float*      v_cache      = (const float*)d_in[2];
  const long long*  slot_tables  = (const long long*)d_in[3];
  const int*        context_lens = (const int*)d_in[4];
  float* out = (float*)d_out;
  float* ws  = (float*)d_ws;   // needs 128*32*4*130*4 B ~= 8.5 MB

  pa_split_kernel<<<dim3(B_ * KVH_, NSPLIT), 32, 0, stream>>>(
      query, k_cache, v_cache, slot_tables, context_lens, ws);
  pa_combine_kernel<<<dim3(B_ * KVH_ * G_), 128, 0, stream>>>(ws, out);
}